// MyModel_61933428415478
// MI455X (gfx1250) — compile-verified
//
#include <hip/hip_runtime.h>

// out[i] = (a+d) + sqrt(|a*d - b*c|) for x[i] = [[a,b],[c,d]] (contiguous float4)
// Streaming problem: 335 MB @ 23.3 TB/s ~= 14 us, memory-bound. Strategy:
//  - one B128 (float4) NT load per matrix per lane, lane-consecutive => coalesced
//  - WAVE-UNIFORM fast/tail split (scalar branch, no exec divergence): interior
//    blocks issue 8 unconditional global_load_b128 back-to-back before any
//    wait -> 8 loads (128 B) in flight per lane for latency hiding
//  - NT hints: 256 MB input > 192 MB L2, single-touch stream both directions
//  - wave32-aware: 256-thread blocks = 8 waves, 8192 blocks

typedef __attribute__((ext_vector_type(4))) float v4f;

#define MATS_PER_THREAD 8
#define BLOCK_THREADS   256
#define MATS_PER_BLOCK  (MATS_PER_THREAD * BLOCK_THREADS)

__device__ __forceinline__ float trace_sqrtdet(v4f m) {
    const float trace = m.x + m.w;
    const float det   = m.x * m.w - m.y * m.z;
    return trace + __builtin_amdgcn_sqrtf(__builtin_fabsf(det));  // |.| is a free src modifier
}

__global__ __launch_bounds__(BLOCK_THREADS) void trace_sqrtdet_kernel(
    const v4f* __restrict__ x, float* __restrict__ out, int n)
{
    const int base = blockIdx.x * MATS_PER_BLOCK + threadIdx.x;

    // Uniform (scalar) predicate: does this whole block fit? -> s_cmp/s_cbranch,
    // fast path executes with full EXEC and no per-lane guards.
    const bool block_full = (unsigned)(blockIdx.x + 1) * (unsigned)MATS_PER_BLOCK
                            <= (unsigned)n;

    if (__builtin_expect(block_full, 1)) {
        v4f m[MATS_PER_THREAD];
#pragma unroll
        for (int k = 0; k < MATS_PER_THREAD; ++k)
            m[k] = __builtin_nontemporal_load(&x[base + k * BLOCK_THREADS]);
#pragma unroll
        for (int k = 0; k < MATS_PER_THREAD; ++k)
            __builtin_nontemporal_store(trace_sqrtdet(m[k]),
                                        &out[base + k * BLOCK_THREADS]);
    } else {
        // Tail path (never taken for N = 16M, kept for generality).
#pragma unroll
        for (int k = 0; k < MATS_PER_THREAD; ++k) {
            const int i = base + k * BLOCK_THREADS;
            if (i < n) {
                const v4f m = __builtin_nontemporal_load(&x[i]);
                __builtin_nontemporal_store(trace_sqrtdet(m), &out[i]);
            }
        }
    }
}

extern "C" void kernel_launch(void* const* d_in, const int* in_sizes, int n_in,
                              void* d_out, int out_size, void* d_ws, size_t ws_size,
                              hipStream_t stream) {
    const v4f* x  = (const v4f*)d_in[0];   // (N, 2, 2) f32 == N contiguous float4
    float*    out = (float*)d_out;         // N f32
    const int n   = out_size;              // number of 2x2 matrices (16,777,216)

    const int blocks = (n + MATS_PER_BLOCK - 1) / MATS_PER_BLOCK;
    trace_sqrtdet_kernel<<<blocks, BLOCK_THREADS, 0, stream>>>(x, out, n);
}